// SpatialEncoding_11579231830078
// MI455X (gfx1250) — compile-verified
//
#include <hip/hip_runtime.h>
#include <cstdint>

// ---------------------------------------------------------------------------
// SpatialEncoding scatter:
//   out = zeros(N,N); out[src[p], dst[p]] = b[min(len[p], maxpd)-1]
// Pure data movement (no matmul content -> no WMMA). Strategy:
//   1) zero-fill with b128 stores (RT hint -> 144MB output stays resident in
//      the 192MB L2, so the 8M random 4B scatter stores are L2 hits).
//   2) persistent-block scatter: stream src/dst/len tiles into LDS with
//      gfx1250 async global->LDS b128 copies, double buffered on ASYNCcnt;
//      consume via ds_load_b128; bias values kept in registers (cndmask
//      select) so the scatter loop's VMEM slots go entirely to stores.
// ---------------------------------------------------------------------------

#define BLOCK 256              // 8 wave32s
#define TILE  2048             // elements per tile per array
#define VEC   4                // dwords per async b128
#define ASYNC_PER_ARRAY (TILE / (BLOCK * VEC))   // 2
#define ASYNC_PER_TILE  (3 * ASYNC_PER_ARRAY)    // 6 async ops / wave / tile

// --- gfx1250 async global->LDS copy (CDNA5 ISA) -------
__device__ __forceinline__ void async_load_b128(unsigned lds_byte_off,
                                                const void* gaddr) {
    // per-lane: copies 16 bytes from the global address gaddr to LDS byte offset lds_byte_off; tracked by ASYNCcnt
    asm volatile("global_load_async_to_lds_b128 %0, %1, off"
                 :
                 : "v"(lds_byte_off), "v"(gaddr)
                 : "memory");
}

__device__ __forceinline__ void wait_asynccnt_le6(void) {
    // wait until ASYNCcnt <= 6: current tile's 6 async ops have completed,
    // the prefetched tile's 6 may still be in flight (in-order completion)
    asm volatile("s_wait_asynccnt 6" ::: "memory");
}

// low 32 bits of a generic pointer to __shared__ == LDS byte offset
__device__ __forceinline__ unsigned lds_offset(const void* p) {
    return (unsigned)(uintptr_t)p;
}

// ---------------------------------------------------------------------------
__global__ void __launch_bounds__(BLOCK)
spatial_zero_kernel(float4* __restrict__ out4, long long n4,
                    float* __restrict__ out, long long n_total) {
    const long long gtid    = (long long)blockIdx.x * blockDim.x + threadIdx.x;
    const long long gstride = (long long)gridDim.x * blockDim.x;
    const float4 z = make_float4(0.f, 0.f, 0.f, 0.f);
    for (long long i = gtid; i < n4; i += gstride) out4[i] = z;
    // scalar remainder (out_size not multiple of 4 — defensive)
    for (long long i = n4 * 4 + gtid; i < n_total; i += gstride) out[i] = 0.f;
}

// ---------------------------------------------------------------------------
__global__ void __launch_bounds__(BLOCK)
spatial_scatter_kernel(const float* __restrict__ bias,
                       const int*   __restrict__ src,
                       const int*   __restrict__ dst,
                       const int*   __restrict__ plen,
                       float*       __restrict__ out,
                       int N, long long P, int maxpd) {
    __shared__ __align__(16) int s_src[2][TILE];
    __shared__ __align__(16) int s_dst[2][TILE];
    __shared__ __align__(16) int s_len[2][TILE];

    const int tid = threadIdx.x;
    const long long numTiles = P / TILE;
    const long long G = (long long)gridDim.x;

    // bias table in registers (uniform scalar loads + cndmask select chain);
    // valid for the reference's maxpd == 5, gather fallback otherwise
    const bool reg_bias = (maxpd <= 5);
    float b0 = 0.f, b1 = 0.f, b2 = 0.f, b3 = 0.f, b4 = 0.f;
    if (reg_bias) {
        b0 = bias[0];
        b1 = (1 < maxpd) ? bias[1] : b0;
        b2 = (2 < maxpd) ? bias[2] : b1;
        b3 = (3 < maxpd) ? bias[3] : b2;
        b4 = (4 < maxpd) ? bias[4] : b3;
    }

    // LDS byte offsets of each staging buffer
    const unsigned lsrc0 = lds_offset(&s_src[0][0]);
    const unsigned ldst0 = lds_offset(&s_dst[0][0]);
    const unsigned llen0 = lds_offset(&s_len[0][0]);
    const unsigned stageBytes = TILE * 4u;

    long long t = blockIdx.x;
    if (t < numTiles) {
        // issue async loads for first tile into stage 0
        {
            const long long base = t * TILE;
#pragma unroll
            for (int r = 0; r < ASYNC_PER_ARRAY; ++r) {
                const unsigned eb = (unsigned)((r * BLOCK + tid) * VEC) * 4u;
                const long long go = base + (long long)(r * BLOCK + tid) * VEC;
                async_load_b128(lsrc0 + eb, src  + go);
                async_load_b128(ldst0 + eb, dst  + go);
                async_load_b128(llen0 + eb, plen + go);
            }
        }
        int st = 0;
        for (; t < numTiles; t += G) {
            // prefetch the next tile into the other stage (clamped so we
            // always issue exactly ASYNC_PER_TILE ops -> fixed-count wait)
            {
                long long nt = t + G;
                long long pf = (nt < numTiles) ? nt : (numTiles - 1);
                const long long base = pf * TILE;
                const unsigned so = (unsigned)(st ^ 1) * stageBytes;
#pragma unroll
                for (int r = 0; r < ASYNC_PER_ARRAY; ++r) {
                    const unsigned eb = so + (unsigned)((r * BLOCK + tid) * VEC) * 4u;
                    const long long go = base + (long long)(r * BLOCK + tid) * VEC;
                    async_load_b128(lsrc0 + eb, src  + go);
                    async_load_b128(ldst0 + eb, dst  + go);
                    async_load_b128(llen0 + eb, plen + go);
                }
            }
            wait_asynccnt_le6();  // current stage's data has landed in LDS
            __syncthreads();      // all 8 waves of the block see it

            // consume current stage: 2 x int4 per array per thread
            // (ds_load_b128, conflict-free), 8 scatter stores per thread
#pragma unroll
            for (int k = 0; k < TILE / (BLOCK * VEC); ++k) {
                const int e = (k * BLOCK + tid) * VEC;
                const int4 sv = *(const int4*)&s_src[st][e];
                const int4 dv = *(const int4*)&s_dst[st][e];
                const int4 lv = *(const int4*)&s_len[st][e];
                const int ss[VEC] = {sv.x, sv.y, sv.z, sv.w};
                const int dd[VEC] = {dv.x, dv.y, dv.z, dv.w};
                const int ll[VEC] = {lv.x, lv.y, lv.z, lv.w};
#pragma unroll
                for (int j = 0; j < VEC; ++j) {
                    int c = (ll[j] < maxpd ? ll[j] : maxpd) - 1;
                    float v;
                    if (reg_bias) {
                        v = b0;
                        v = (c == 1) ? b1 : v;
                        v = (c == 2) ? b2 : v;
                        v = (c == 3) ? b3 : v;
                        v = (c == 4) ? b4 : v;
                    } else {
                        v = bias[c];
                    }
                    out[(size_t)ss[j] * (size_t)N + (size_t)dd[j]] = v;
                }
            }
            __syncthreads();      // buffers free before next iteration's issue
            st ^= 1;
        }
    }

    // tail: P mod TILE elements, direct loads
    const long long gtid    = (long long)blockIdx.x * BLOCK + tid;
    const long long gstride = G * BLOCK;
    for (long long i = numTiles * TILE + gtid; i < P; i += gstride) {
        const int s = src[i];
        const int d = dst[i];
        int l = plen[i];
        int c = (l < maxpd ? l : maxpd) - 1;
        float v;
        if (reg_bias) {
            v = b0;
            v = (c == 1) ? b1 : v;
            v = (c == 2) ? b2 : v;
            v = (c == 3) ? b3 : v;
            v = (c == 4) ? b4 : v;
        } else {
            v = bias[c];
        }
        out[(size_t)s * (size_t)N + (size_t)d] = v;
    }
}

// ---------------------------------------------------------------------------
extern "C" void kernel_launch(void* const* d_in, const int* in_sizes, int n_in,
                              void* d_out, int out_size, void* d_ws, size_t ws_size,
                              hipStream_t stream) {
    (void)d_ws; (void)ws_size; (void)n_in;
    // inputs (setup_inputs order): x, b, src_idx, dst_idx, path_len
    const float* bias = (const float*)d_in[1];
    const int*   src  = (const int*)d_in[2];
    const int*   dst  = (const int*)d_in[3];
    const int*   plen = (const int*)d_in[4];
    float* out = (float*)d_out;

    const int       maxpd = in_sizes[1];            // 5
    const long long P     = (long long)in_sizes[2]; // 8M
    const int       N     = in_sizes[0] / 128;      // D_FEAT = 128 -> N = 6000

    // 1) zero the 144MB output (leaves it resident in the 192MB L2)
    const long long n_total = (long long)out_size;
    const long long n4      = n_total / 4;
    spatial_zero_kernel<<<4096, BLOCK, 0, stream>>>((float4*)out, n4, out, n_total);

    // 2) async-pipelined scatter
    spatial_scatter_kernel<<<1024, BLOCK, 0, stream>>>(bias, src, dst, plen,
                                                       out, N, P, maxpd);
}